// FeatureSim_77584289235171
// MI455X (gfx1250) — compile-verified
//
#include <hip/hip_runtime.h>
#include <stdint.h>

typedef __attribute__((ext_vector_type(2))) float v2f;
typedef __attribute__((ext_vector_type(8))) float v8f;

constexpr int kB     = 8;
constexpr int kN     = 1024;
constexpr int kD     = 64;
constexpr int kNF    = 11;          // NUM_FEAT
constexpr int kFS    = 13;          // LDS feature stride (odd -> conflict-free)
constexpr int kWaves = 8;           // waves per block, each owns a column strip
constexpr int kThr   = kWaves * 32; // 256 threads
constexpr int kStrip = kN / kWaves; // 128 columns per wave

__global__ __launch_bounds__(kThr) void feature_sim_softmax_kernel(
    const float* __restrict__ x,
    const int*   __restrict__ x_lengths,
    const float* __restrict__ w,
    float*       __restrict__ out)
{
    __shared__ float s_feat[kN * kFS];        // all keys' features (~52 KB)
    __shared__ float s_partial[kWaves][16];   // per-wave strip row sums
    __shared__ float s_rowsum[16];            // combined row sums

    const int b    = blockIdx.y;
    const int i0   = blockIdx.x * 16;         // this block's 16-row query tile
    const int tid  = threadIdx.x;
    const int lane = tid & 31;
    const int wv   = tid >> 5;
    const int len  = x_lengths[b];

    // ---- Stage key features via CDNA5 async global->LDS copies (ASYNCcnt) ----
    // 12 x B32 per key row (11 used + 1 pad); LDS offset = low 32 bits of the
    // generic shared pointer (flat aperture keeps LDS offset in addr[31:0]).
    for (int j = tid; j < kN; j += kThr) {
        const float* src = x + ((size_t)b * kN + j) * kD;
        const unsigned ldsBase = (unsigned)(uintptr_t)&s_feat[j * kFS];
#pragma unroll
        for (int k = 0; k < 12; ++k) {
            const unsigned l = ldsBase + 4u * k;
            const float*   g = src + k;
            asm volatile("global_load_async_to_lds_b32 %0, %1, off"
                         :: "v"(l), "v"(g) : "memory");
        }
    }

    float wk[kNF];
#pragma unroll
    for (int k = 0; k < kNF; ++k) wk[k] = w[k];

    asm volatile("s_wait_asynccnt 0" ::: "memory");
    __syncthreads();

    // ---- Pass 1: strip row-sums of exp(score) on the matrix pipe ----
    // A fragment of V_WMMA_F32_16X16X4_F32: lane m in [0,16) holds row m;
    // lanes 0-15 -> K=0,1 (VGPR0,1); lanes 16-31 -> K=2,3.  B = ones(4x16)
    // (layout-proof), so D[m,*] accumulates exact f32 row sums.
    const int m     = lane & 15;
    const int half  = lane >> 4;
    const int row   = i0 + m;
    const int jbase = wv * kStrip;

    float xi[kNF];
#pragma unroll
    for (int k = 0; k < kNF; ++k) xi[k] = s_feat[row * kFS + k];

    v8f acc = {};
    v2f bones; bones[0] = 1.0f; bones[1] = 1.0f;

    for (int jc = jbase; jc < jbase + kStrip; jc += 4) {
        const int ja = jc + half * 2;
        v2f a;
#pragma unroll
        for (int t = 0; t < 2; ++t) {
            const int j = ja + t;
            float s = 0.0f;
#pragma unroll
            for (int k = 0; k < kNF; ++k)
                s = fmaf(wk[k], fabsf(xi[k] - s_feat[j * kFS + k]), s);
            const float val = (s < 1.0f) ? -s : 0.0f;   // scores in (-1, 0]
            a[t] = (j < len) ? __expf(val) : 0.0f;      // invalid keys -> 0
        }
        acc = __builtin_amdgcn_wmma_f32_16x16x4_f32(
            false, a, false, bones, (short)0, acc, false, false);
    }

    // C/D layout: VGPR v holds row v (lanes 0-15) / row v+8 (lanes 16-31);
    // all lanes of a half carry identical sums -> publish from lane 0 / 16.
    if (lane == 0) {
#pragma unroll
        for (int v = 0; v < 8; ++v) s_partial[wv][v] = acc[v];
    }
    if (lane == 16) {
#pragma unroll
        for (int v = 0; v < 8; ++v) s_partial[wv][8 + v] = acc[v];
    }
    __syncthreads();

    // Combine the 8 strip partials per row.
    if (tid < 16) {
        float t = 0.0f;
#pragma unroll
        for (int v = 0; v < kWaves; ++v) t += s_partial[v][tid];
        s_rowsum[tid] = t;                  // sum >= e^-1, never zero
    }
    __syncthreads();

    // ---- Pass 2: recompute exp (cheap), scale, stream out own strip ----
    for (int mm = 0; mm < 16; ++mm) {
        const int   r    = i0 + mm;
        const float rinv = 1.0f / s_rowsum[mm];
        float xr[kNF];
#pragma unroll
        for (int k = 0; k < kNF; ++k) xr[k] = s_feat[r * kFS + k];
        float* orow = out + ((size_t)b * kN + r) * kN;
        for (int j = jbase + lane; j < jbase + kStrip; j += 32) {
            float s = 0.0f;
#pragma unroll
            for (int k = 0; k < kNF; ++k)
                s = fmaf(wk[k], fabsf(xr[k] - s_feat[j * kFS + k]), s);
            const float val = (s < 1.0f) ? -s : 0.0f;
            const float e   = (j < len) ? __expf(val) : 0.0f;
            __builtin_nontemporal_store(e * rinv, &orow[j]);  // TH=NT stream
        }
    }
}

extern "C" void kernel_launch(void* const* d_in, const int* in_sizes, int n_in,
                              void* d_out, int out_size, void* d_ws, size_t ws_size,
                              hipStream_t stream) {
    const float* x  = (const float*)d_in[0];
    const int*   xl = (const int*)d_in[1];
    const float* fi = (const float*)d_in[2];
    dim3 grid(kN / 16, kB);   // 64 x 8 = 512 blocks, 8 waves each = 4096 waves
    dim3 block(kThr);
    feature_sim_softmax_kernel<<<grid, block, 0, stream>>>(x, xl, fi, (float*)d_out);
}